// MyGaussianPDF_1391569403986
// MI455X (gfx1250) — compile-verified
//
#include <hip/hip_runtime.h>

// Gaussian PDF: out[i] = exp(-0.5 * 0.54 * ||x_i - mu||^2), N = 16,777,216.
// Pure HBM-bound stream (12 B/elem, ~0.6 flop/byte): 201 MB @ 23.3 TB/s ~= 8.6 us.
// CDNA5 async global->LDS DMA (ASYNCcnt) double-buffer + NT b128 loads/stores.

#define THREADS        256
#define PTS_PER_THREAD 4
#define TILE_PTS       (THREADS * PTS_PER_THREAD)   // 1024 points / tile
#define TILE_FLOATS    (TILE_PTS * 2)               // 2048 floats = 8 KB / tile

typedef int   v4i __attribute__((ext_vector_type(4)));
typedef float v4f __attribute__((ext_vector_type(4)));

#if defined(__HIP_DEVICE_COMPILE__) && defined(__gfx1250__) &&                 \
    __has_builtin(__builtin_amdgcn_global_load_async_to_lds_b128) &&           \
    __has_builtin(__builtin_amdgcn_s_wait_asynccnt)
#define USE_ASYNC 1
#else
#define USE_ASYNC 0
#endif

// CPol: TH[2:0] = 1 -> non-temporal (single-pass stream, no L2 reuse).
#define CPOL_NT 1

#if USE_ASYNC
typedef v4i __attribute__((address_space(1))) g_v4i;  // global int4
typedef v4i __attribute__((address_space(3))) l_v4i;  // LDS int4

static __device__ __forceinline__ g_v4i* to_glob(const void* p) {
  return (g_v4i*)(uintptr_t)p;
}
static __device__ __forceinline__ l_v4i* to_lds(void* p) {
  // Flat LDS address: low 32 bits are the LDS byte offset (CDNA5 aperture rule).
  return (l_v4i*)(unsigned int)(uintptr_t)p;
}
#endif

static __device__ __forceinline__ v4f gauss8(v4f a, v4f b, float mux, float muy) {
  v4f r;
  float dx, dy, s;
  dx = a.x - mux; dy = a.y - muy; s = dx * dx + dy * dy;
  r.x = __expf(-0.5f * (s * 0.54f));
  dx = a.z - mux; dy = a.w - muy; s = dx * dx + dy * dy;
  r.y = __expf(-0.5f * (s * 0.54f));
  dx = b.x - mux; dy = b.y - muy; s = dx * dx + dy * dy;
  r.z = __expf(-0.5f * (s * 0.54f));
  dx = b.z - mux; dy = b.w - muy; s = dx * dx + dy * dy;
  r.w = __expf(-0.5f * (s * 0.54f));
  return r;
}

__global__ __launch_bounds__(THREADS) void gauss_main(
    const float* __restrict__ x, const float* __restrict__ mu,
    float* __restrict__ out, unsigned ntiles) {
  const float mux = mu[0];
  const float muy = mu[1];
  const unsigned tid    = threadIdx.x;
  unsigned       t      = blockIdx.x;
  const unsigned stride = gridDim.x;

#if USE_ASYNC
  __shared__ __align__(16) float lds[2][TILE_FLOATS];  // 16 KB double buffer
  const unsigned wave  = tid >> 5;
  const unsigned lane  = tid & 31u;
  const unsigned chunk = wave * 1024u + lane * 16u;  // this lane's 16B piece

  auto issue = [&](unsigned tile, unsigned buf) {
    // Each wave DMAs its contiguous 1 KB slice of the tile as 2x 512B b128 ops.
    const char* g = (const char*)x + (size_t)tile * (TILE_FLOATS * 4u) + chunk;
    char*       l = (char*)&lds[buf][0] + chunk;
    __builtin_amdgcn_global_load_async_to_lds_b128(to_glob(g), to_lds(l), 0, CPOL_NT);
    __builtin_amdgcn_global_load_async_to_lds_b128(to_glob(g + 512), to_lds(l + 512), 0, CPOL_NT);
  };

  unsigned cur = 0;
  if (t < ntiles) issue(t, 0);
  for (; t < ntiles; t += stride) {
    const unsigned nxt = t + stride;
    if (nxt < ntiles) {
      issue(nxt, cur ^ 1u);                 // prefetch next tile into other buffer
      __builtin_amdgcn_s_wait_asynccnt(2);  // current tile's 2 DMAs done
    } else {
      __builtin_amdgcn_s_wait_asynccnt(0);
    }
    asm volatile("" ::: "memory");  // keep LDS reads after the wait

    const v4f a = *(const v4f*)&lds[cur][tid * 8u];
    const v4f b = *(const v4f*)&lds[cur][tid * 8u + 4u];
    const v4f r = gauss8(a, b, mux, muy);
    __builtin_nontemporal_store(r, (v4f*)(out + (size_t)t * TILE_PTS + tid * 4u));
    cur ^= 1u;
  }
#else
  // Fallback: direct NT b128 streaming (same tiling / launch geometry).
  for (; t < ntiles; t += stride) {
    const v4f* xq = (const v4f*)(x + (size_t)t * TILE_FLOATS) + tid * 2u;
    const v4f a = __builtin_nontemporal_load(xq);
    const v4f b = __builtin_nontemporal_load(xq + 1);
    const v4f r = gauss8(a, b, mux, muy);
    __builtin_nontemporal_store(r, (v4f*)(out + (size_t)t * TILE_PTS + tid * 4u));
  }
#endif
}

// Scalar tail for n % TILE_PTS != 0 (not hit for N = 16M, kept for correctness).
__global__ void gauss_tail(const float* __restrict__ x, const float* __restrict__ mu,
                           float* __restrict__ out, long long start, long long n) {
  long long i = start + (long long)blockIdx.x * blockDim.x + threadIdx.x;
  if (i < n) {
    const float dx = x[2 * i]     - mu[0];
    const float dy = x[2 * i + 1] - mu[1];
    const float s  = dx * dx + dy * dy;
    out[i] = __expf(-0.5f * (s * 0.54f));
  }
}

extern "C" void kernel_launch(void* const* d_in, const int* in_sizes, int n_in,
                              void* d_out, int out_size, void* d_ws, size_t ws_size,
                              hipStream_t stream) {
  (void)n_in; (void)out_size; (void)d_ws; (void)ws_size;
  const float* x   = (const float*)d_in[0];
  const float* mu  = (const float*)d_in[1];
  float*       out = (float*)d_out;

  const long long npts   = (long long)in_sizes[0] / 2;  // in_sizes[0] = N*2 floats
  const unsigned  ntiles = (unsigned)(npts / TILE_PTS);

  if (ntiles > 0) {
    unsigned blocks = ntiles < 4096u ? ntiles : 4096u;  // persistent, 2-4 tiles/block
    gauss_main<<<blocks, THREADS, 0, stream>>>(x, mu, out, ntiles);
  }
  const long long start = (long long)ntiles * TILE_PTS;
  const long long rem   = npts - start;
  if (rem > 0) {
    unsigned b = (unsigned)((rem + 255) / 256);
    gauss_tail<<<b, 256, 0, stream>>>(x, mu, out, start, npts);
  }
}